// AttnDecoderRNN_46583215292925
// MI455X (gfx1250) — compile-verified
//
#include <hip/hip_runtime.h>
#include <hip/hip_bf16.h>

#define B_    128
#define S_    2048
#define H_    256
#define E_    256
#define VOCAB 32000

typedef __attribute__((ext_vector_type(2))) float v2f;
typedef __attribute__((ext_vector_type(8))) float v8f;

// branch-free fast tanh / sigmoid (v_exp_f32 + v_rcp_f32, no exec-mask churn)
__device__ __forceinline__ float tanh_fast(float x) {
    float e = __expf(2.0f * x);
    return 1.0f - 2.0f * __builtin_amdgcn_rcpf(e + 1.0f);
}
__device__ __forceinline__ float sigmoid_fast(float x) {
    return __builtin_amdgcn_rcpf(1.0f + __expf(-x));
}

// ---------------- w2h = hidden[1] @ W2 + b2 : [128,256] ----------------
__global__ __launch_bounds__(256) void k_w2h(const float* __restrict__ hidden,
                                             const float* __restrict__ W2,
                                             const float* __restrict__ b2,
                                             float* __restrict__ w2h) {
    int b = blockIdx.x, n = threadIdx.x;
    const float* hrow = hidden + (size_t)(1 * B_ + b) * H_;
    float acc = b2[n];
#pragma unroll 4
    for (int h = 0; h < H_; ++h) acc += hrow[h] * W2[h * H_ + n];
    w2h[b * H_ + n] = acc;
}

// ---- pack W1 so each WMMA B-fragment pair is one contiguous b64 per lane ----
// W1p[k0*H + n*4 + o] = W1[(k0+o)*H + n],  k0 multiple of 4, o in 0..3
__global__ __launch_bounds__(256) void k_packW1(const float* __restrict__ W1,
                                                float* __restrict__ W1p) {
    int k = blockIdx.x, n = threadIdx.x;
    W1p[(size_t)(k >> 2) * (H_ * 4) + n * 4 + (k & 3)] = W1[(size_t)k * H_ + n];
}

// ---- fused attention scores: scores[b,s] = V . tanh(enc[b,s,:]@W1 + w2h[b,:]) ----
// One wave per (b, 16-row s-tile); A staged in LDS (b128 fill, pad-260 stride);
// 16 N-tiles x 64 K-steps of v_wmma_f32_16x16x4_f32 per wave.
__global__ __launch_bounds__(32) void k_attn_scores(const float* __restrict__ enc,
                                                    const float* __restrict__ W1p,
                                                    const float* __restrict__ w2h,
                                                    const float* __restrict__ V,
                                                    float* __restrict__ scores) {
    __shared__ __align__(16) float lds[16 * 260];  // stride 260 words: b128-aligned, bank-clean
    int b = blockIdx.y, s0 = blockIdx.x * 16;
    int lane = threadIdx.x;
    const float* encb = enc + ((size_t)b * S_ + s0) * H_;
    for (int i = lane; i < 16 * (H_ / 4); i += 32) {     // 32 b128 iters
        int r = i >> 6, c4 = (i & 63) * 4;
        *(float4*)&lds[r * 260 + c4] = *(const float4*)(encb + (size_t)r * H_ + c4);
    }
    __syncthreads();

    int ln = lane & 15;
    int hi = lane >> 4;               // 0: K=k0,k0+1 ; 1: K=k0+2,k0+3
    const float* w2b = w2h + b * H_;
    float part[8];
#pragma unroll
    for (int j = 0; j < 8; ++j) part[j] = 0.0f;

    for (int nt = 0; nt < 16; ++nt) {
        int n0 = nt * 16;
        const float* bp = W1p + ((n0 + ln) << 2) + 2 * hi;
        v8f acc = {0.f, 0.f, 0.f, 0.f, 0.f, 0.f, 0.f, 0.f};
#pragma unroll 8
        for (int k0 = 0; k0 < H_; k0 += 4) {
            int ka = k0 + 2 * hi;
            v2f a  = *(const v2f*)&lds[ln * 260 + ka];  // ds b64
            v2f bb = *(const v2f*)(bp + k0 * H_);       // global b64, coalesced
            acc = __builtin_amdgcn_wmma_f32_16x16x4_f32(false, a, false, bb,
                                                        (short)0, acc, false, false);
        }
        int   col = n0 + ln;
        float w2  = w2b[col];
        float vv  = V[col];
#pragma unroll
        for (int j = 0; j < 8; ++j)
            part[j] += tanh_fast(acc[j] + w2) * vv;
    }
    // reduce over the 16 lanes of each half (xor masks 1..8 stay within halves)
#pragma unroll
    for (int j = 0; j < 8; ++j) {
        float p = part[j];
        p += __shfl_xor(p, 1, 32);
        p += __shfl_xor(p, 2, 32);
        p += __shfl_xor(p, 4, 32);
        p += __shfl_xor(p, 8, 32);
        part[j] = p;
    }
    if (ln == 0) {
#pragma unroll
        for (int j = 0; j < 8; ++j)
            scores[(size_t)b * S_ + s0 + j + 8 * hi] = part[j];
    }
}

// ---------------- softmax over S (in place) ----------------
__global__ __launch_bounds__(256) void k_softmax_s(float* __restrict__ sc) {
    __shared__ float red[256];
    int b = blockIdx.x, t = threadIdx.x;
    float* row = sc + (size_t)b * S_;
    float m = -__builtin_inff();
    for (int s = t; s < S_; s += 256) m = fmaxf(m, row[s]);
    red[t] = m; __syncthreads();
    for (int o = 128; o; o >>= 1) { if (t < o) red[t] = fmaxf(red[t], red[t + o]); __syncthreads(); }
    m = red[0]; __syncthreads();
    float sum = 0.f;
    for (int s = t; s < S_; s += 256) { float e = __expf(row[s] - m); row[s] = e; sum += e; }
    red[t] = sum; __syncthreads();
    for (int o = 128; o; o >>= 1) { if (t < o) red[t] += red[t + o]; __syncthreads(); }
    float inv = __builtin_amdgcn_rcpf(red[0]);
    for (int s = t; s < S_; s += 256) row[s] *= inv;
}

// ---- Xa[b,h] = sum_s a[b,s]*enc[b,s,h] : b128 streaming, 4 s-stripes/block ----
__global__ __launch_bounds__(256) void k_xa(const float* __restrict__ a,
                                            const float* __restrict__ enc,
                                            float* __restrict__ Xa) {
    __shared__ __align__(16) float red[4][H_];
    int b  = blockIdx.x;
    int t  = threadIdx.x;
    int hx = (t & 63) * 4;          // 64 threads cover H with float4
    int sy = t >> 6;                // 4 s-stripes
    const float* ab = a + (size_t)b * S_;
    float ax = 0.f, ay = 0.f, az = 0.f, aw = 0.f;
    for (int s = sy; s < S_; s += 4) {
        float  w = ab[s];
        float4 e = *(const float4*)(enc + ((size_t)b * S_ + s) * H_ + hx);
        ax += w * e.x; ay += w * e.y; az += w * e.z; aw += w * e.w;
    }
    red[sy][hx]     = ax;
    red[sy][hx + 1] = ay;
    red[sy][hx + 2] = az;
    red[sy][hx + 3] = aw;
    __syncthreads();
    if (sy == 0) {
#pragma unroll
        for (int c = 0; c < 4; ++c) {
            int h = hx + c;
            Xa[b * H_ + h] = red[0][h] + red[1][h] + red[2][h] + red[3][h];
        }
    }
}

// ---------------- res = concat(emb[inp], Xa) @ W3 + b3 ----------------
__global__ __launch_bounds__(256) void k_res(const int* __restrict__ inp,
                                             const float* __restrict__ emb,
                                             const float* __restrict__ Xa,
                                             const float* __restrict__ W3,
                                             const float* __restrict__ b3,
                                             float* __restrict__ res) {
    __shared__ float xe[E_], xa[H_];
    int b = blockIdx.x, n = threadIdx.x;
    xe[n] = emb[(size_t)inp[b] * E_ + n];
    xa[n] = Xa[b * H_ + n];
    __syncthreads();
    float acc = b3[n];
#pragma unroll 4
    for (int k = 0; k < E_; ++k) acc += xe[k] * W3[k * H_ + n];
#pragma unroll 4
    for (int k = 0; k < H_; ++k) acc += xa[k] * W3[(E_ + k) * H_ + n];
    res[b * H_ + n] = acc;
}

// ---------------- GRU cell (PyTorch gate order r,z,n), b128 weight walks ----------------
__global__ __launch_bounds__(256) void k_gru(const float* __restrict__ x,
                                             const float* __restrict__ hprev,
                                             const float* __restrict__ Wih,
                                             const float* __restrict__ Whh,
                                             const float* __restrict__ bih,
                                             const float* __restrict__ bhh,
                                             float* __restrict__ hout) {
    __shared__ __align__(16) float xs[H_], hs[H_];
    int b = blockIdx.x, h = threadIdx.x;
    xs[h] = x[b * H_ + h];
    hs[h] = hprev[b * H_ + h];
    __syncthreads();
    float ir = bih[h], iz = bih[H_ + h], in_ = bih[2 * H_ + h];
    float hr = bhh[h], hz = bhh[H_ + h], hn  = bhh[2 * H_ + h];
    const float* wr = Wih + (size_t)h * H_;
    const float* wz = Wih + (size_t)(H_ + h) * H_;
    const float* wn = Wih + (size_t)(2 * H_ + h) * H_;
    const float* vr = Whh + (size_t)h * H_;
    const float* vz = Whh + (size_t)(H_ + h) * H_;
    const float* vn = Whh + (size_t)(2 * H_ + h) * H_;
#pragma unroll 2
    for (int k = 0; k < H_; k += 4) {
        float4 xk = *(const float4*)(xs + k);
        float4 hk = *(const float4*)(hs + k);
        float4 w;
        w = *(const float4*)(wr + k); ir += xk.x*w.x + xk.y*w.y + xk.z*w.z + xk.w*w.w;
        w = *(const float4*)(wz + k); iz += xk.x*w.x + xk.y*w.y + xk.z*w.z + xk.w*w.w;
        w = *(const float4*)(wn + k); in_ += xk.x*w.x + xk.y*w.y + xk.z*w.z + xk.w*w.w;
        w = *(const float4*)(vr + k); hr += hk.x*w.x + hk.y*w.y + hk.z*w.z + hk.w*w.w;
        w = *(const float4*)(vz + k); hz += hk.x*w.x + hk.y*w.y + hk.z*w.z + hk.w*w.w;
        w = *(const float4*)(vn + k); hn += hk.x*w.x + hk.y*w.y + hk.z*w.z + hk.w*w.w;
    }
    float r = sigmoid_fast(ir + hr);
    float z = sigmoid_fast(iz + hz);
    float n = tanh_fast(in_ + r * hn);
    hout[b * H_ + h] = (1.f - z) * n + z * hs[h];
}

// ------- logits = h1 @ Wout.T + bout, via f32 WMMA (M=128, N=32000, K=256) -------
__global__ __launch_bounds__(32) void k_logits(const float* __restrict__ h1,
                                               const float* __restrict__ Wout,
                                               const float* __restrict__ bout,
                                               float* __restrict__ out) {
    int n0 = blockIdx.x * 16;         // vocab tile
    int m0 = blockIdx.y * 16;         // batch tile
    int lane = threadIdx.x, ln = lane & 15, hi = lane >> 4;
    const float* arow = h1   + (size_t)(m0 + ln) * H_ + 2 * hi;
    const float* brow = Wout + (size_t)(n0 + ln) * H_ + 2 * hi;  // B^T: contiguous pair
    v8f acc = {0.f, 0.f, 0.f, 0.f, 0.f, 0.f, 0.f, 0.f};
#pragma unroll 8
    for (int k0 = 0; k0 < H_; k0 += 4) {
        v2f a  = *(const v2f*)(arow + k0);   // global b64
        v2f bb = *(const v2f*)(brow + k0);   // global b64
        acc = __builtin_amdgcn_wmma_f32_16x16x4_f32(false, a, false, bb,
                                                    (short)0, acc, false, false);
    }
    float bo = bout[n0 + ln];
#pragma unroll
    for (int j = 0; j < 8; ++j)
        out[(size_t)(m0 + j + 8 * hi) * VOCAB + n0 + ln] = acc[j] + bo;
}

// ---------------- log_softmax over vocab: stats then in-place apply ----------------
__global__ __launch_bounds__(256) void k_lsm_stats(const float* __restrict__ out,
                                                   float* __restrict__ rmax,
                                                   float* __restrict__ rlse) {
    __shared__ float red[256];
    int b = blockIdx.x, t = threadIdx.x;
    const float* row = out + (size_t)b * VOCAB;
    float m = -__builtin_inff();
    for (int v = t; v < VOCAB; v += 256) m = fmaxf(m, row[v]);
    red[t] = m; __syncthreads();
    for (int o = 128; o; o >>= 1) { if (t < o) red[t] = fmaxf(red[t], red[t + o]); __syncthreads(); }
    m = red[0]; __syncthreads();
    float s = 0.f;
    for (int v = t; v < VOCAB; v += 256) s += __expf(row[v] - m);
    red[t] = s; __syncthreads();
    for (int o = 128; o; o >>= 1) { if (t < o) red[t] += red[t + o]; __syncthreads(); }
    if (t == 0) { rmax[b] = m; rlse[b] = logf(red[0]); }
}

__global__ __launch_bounds__(256) void k_lsm_apply(float* __restrict__ out,
                                                   const float* __restrict__ rmax,
                                                   const float* __restrict__ rlse) {
    size_t i = (size_t)blockIdx.x * 256 + threadIdx.x;
    int b = (int)(i / VOCAB);
    out[i] = out[i] - rmax[b] - rlse[b];
}

extern "C" void kernel_launch(void* const* d_in, const int* in_sizes, int n_in,
                              void* d_out, int out_size, void* d_ws, size_t ws_size,
                              hipStream_t stream) {
    const int*   inp    = (const int*)  d_in[0];
    const float* hidden = (const float*)d_in[1];
    const float* enc    = (const float*)d_in[2];
    const float* emb    = (const float*)d_in[3];
    const float* W1     = (const float*)d_in[4];
    const float* W2     = (const float*)d_in[5];
    const float* W3     = (const float*)d_in[6];
    const float* b2     = (const float*)d_in[7];
    const float* b3     = (const float*)d_in[8];
    const float* V      = (const float*)d_in[9];
    const float* Wih0   = (const float*)d_in[10];
    const float* Whh0   = (const float*)d_in[11];
    const float* bih0   = (const float*)d_in[12];
    const float* bhh0   = (const float*)d_in[13];
    const float* Wih1   = (const float*)d_in[14];
    const float* Whh1   = (const float*)d_in[15];
    const float* bih1   = (const float*)d_in[16];
    const float* bhh1   = (const float*)d_in[17];
    const float* Wout   = (const float*)d_in[18];
    const float* bout   = (const float*)d_in[19];

    float* out = (float*)d_out;                 // [128, 32000] log-probs
    float* h0  = out + (size_t)B_ * VOCAB;      // new_hidden[0]
    float* h1  = h0 + B_ * H_;                  // new_hidden[1]

    float* ws   = (float*)d_ws;
    float* w2h  = ws;                     // 128*256
    float* sc   = w2h + B_ * H_;          // 128*2048 (scores -> softmax in place)
    float* Xa   = sc + (size_t)B_ * S_;   // 128*256
    float* res  = Xa + B_ * H_;           // 128*256
    float* rmax = res + B_ * H_;          // 128
    float* rlse = rmax + B_;              // 128
    float* W1p  = rlse + B_;              // 256*256 packed W1

    k_w2h        <<<dim3(B_),          dim3(H_),  0, stream>>>(hidden, W2, b2, w2h);
    k_packW1     <<<dim3(H_),          dim3(H_),  0, stream>>>(W1, W1p);
    k_attn_scores<<<dim3(S_ / 16, B_), dim3(32),  0, stream>>>(enc, W1p, w2h, V, sc);
    k_softmax_s  <<<dim3(B_),          dim3(256), 0, stream>>>(sc);
    k_xa         <<<dim3(B_),          dim3(256), 0, stream>>>(sc, enc, Xa);
    k_res        <<<dim3(B_),          dim3(H_),  0, stream>>>(inp, emb, Xa, W3, b3, res);
    k_gru        <<<dim3(B_),          dim3(H_),  0, stream>>>(res, hidden,           Wih0, Whh0, bih0, bhh0, h0);
    k_gru        <<<dim3(B_),          dim3(H_),  0, stream>>>(h0,  hidden + B_ * H_, Wih1, Whh1, bih1, bhh1, h1);
    k_logits     <<<dim3(VOCAB / 16, B_ / 16), dim3(32), 0, stream>>>(h1, Wout, bout, out);
    k_lsm_stats  <<<dim3(B_),          dim3(256), 0, stream>>>(out, rmax, rlse);
    k_lsm_apply  <<<dim3((B_ * VOCAB) / 256), dim3(256), 0, stream>>>(out, rmax, rlse);
}